// QuantumConvolutionLayer_22196390986564
// MI455X (gfx1250) — compile-verified
//
#include <hip/hip_runtime.h>
#include <stdint.h>

typedef __attribute__((ext_vector_type(2))) float v2f;
typedef __attribute__((ext_vector_type(8))) float v8f;

#define PI_HALF 1.57079632679489662f

// One block = one (b, h) output row: 128 sites, 2 outputs each.
// 8 waves x 16 sites/wave via v_wmma_f32_16x16x4_f32.
__global__ __launch_bounds__(256) void qconv_wmma_kernel(
    const float* __restrict__ x,     // (256,1,256,256)
    const float* __restrict__ Ure,   // (4,4)
    const float* __restrict__ Uim,   // (4,4)
    float* __restrict__ out)         // (256,2,128,128)
{
    __shared__ float row[256];       // staged input row (1 KB)
    __shared__ float zbuf[2][128];   // staged outputs for coalesced store

    const int tid = threadIdx.x;
    const int h   = blockIdx.x;      // 0..127
    const int b   = blockIdx.y;      // 0..255

    const int J = min(2 * h + 1, 254);
    const float* rowBase = x + ((size_t)b * 256 + (size_t)J) * 256;

    // --- CDNA5 async load: global row -> LDS (ASYNCcnt path) ---
    {
        unsigned      ldsOff = (unsigned)(uintptr_t)(void*)&row[tid];
        const float*  g      = rowBase + tid;
        asm volatile("global_load_async_to_lds_b32 %0, %1, off"
                     :: "v"(ldsOff), "v"(g) : "memory");
        asm volatile("s_wait_asynccnt 0x0" ::: "memory");
    }
    __syncthreads();

    const int lane = tid & 31;       // wave32
    const int wave = tid >> 5;       // 0..7
    const int half = lane >> 4;      // 0: A holds K=0,1 ; 1: A holds K=2,3
    const int n    = lane & 15;      // matrix M (for A) / N (for B,D) index
    const int base = wave * 16;      // first site of this wave

    // Build A (16x4 f32 psi rows). Lane L and L+16 both serve site base+ (L&15).
    const int   p    = base + n;
    const int   col0 = min(2 * p + 1, 254);
    const float phi0 = row[col0];
    const float phi1 = row[col0 + 1];
    float sa, ca, sb, cb;
    __sincosf(PI_HALF * phi0, &sa, &ca);
    __sincosf(PI_HALF * phi1, &sb, &cb);
    const float f0 = half ? sa : ca;        // cos(a) rows feed K=0,1 ; sin(a) rows feed K=2,3
    v2f A;
    A.x = f0 * cb;                          // psi K = 0 (or 2)
    A.y = f0 * sb;                          // psi K = 1 (or 3)

    // Build B (4x16): cols 0..3 = Ure^T, cols 4..7 = Uim^T, rest 0.
    // Lanes 0-15 carry rows K=0,1 in v0,v1; lanes 16-31 carry rows K=2,3.
    const int k0 = half * 2;
    float b0 = 0.0f, b1 = 0.0f;
    if (n < 4)      { b0 = Ure[n * 4 + k0];       b1 = Ure[n * 4 + k0 + 1]; }
    else if (n < 8) { b0 = Uim[(n - 4) * 4 + k0]; b1 = Uim[(n - 4) * 4 + k0 + 1]; }
    v2f B;
    B.x = b0;
    B.y = b1;

    // D[m][0:4] = pr, D[m][4:8] = pi for 16 sites in one shot.
    v8f C = {};
    v8f D = __builtin_amdgcn_wmma_f32_16x16x4_f32(
        /*neg_a=*/false, A, /*neg_b=*/false, B,
        /*c_mod=*/(short)0, C, /*reuse_a=*/false, /*reuse_b=*/false);

    // probs = pr^2 + pi^2 ; z0 = p0+p1-p2-p3 ; z1 = p0-p1+p2-p3
    // D-layout: VGPR r holds M=r (lanes 0-15, N=lane) and M=8+r (lanes 16-31, N=lane-16).
    #pragma unroll
    for (int r = 0; r < 8; ++r) {
        float t   = D[r] * D[r];
        float pr  = t + __shfl_xor(t, 4, 32);    // probs[N] at lanes N (N<4), per half
        float r1  = __shfl_xor(pr, 1, 32);
        float s01 = pr + r1;                     // lane0: p0+p1, lane2: p2+p3
        float d01 = pr - r1;                     // lane0: p0-p1, lane2: p2-p3
        float z0  = s01 - __shfl_xor(s01, 2, 32);
        float z1  = d01 + __shfl_xor(d01, 2, 32);
        if (n == 0) {                            // valid at lanes 0 and 16
            const int m = base + half * 8 + r;
            zbuf[0][m] = z0;
            zbuf[1][m] = z1;
        }
    }
    __syncthreads();

    // Coalesced store: two contiguous 512B rows.
    const int c = tid >> 7;        // 0,1 -> output channel
    const int w = tid & 127;
    out[(((size_t)b * 2 + c) * 128 + h) * 128 + w] = zbuf[c][w];
}

extern "C" void kernel_launch(void* const* d_in, const int* in_sizes, int n_in,
                              void* d_out, int out_size, void* d_ws, size_t ws_size,
                              hipStream_t stream) {
    const float* x   = (const float*)d_in[0];
    const float* Ure = (const float*)d_in[1];
    const float* Uim = (const float*)d_in[2];
    float* out = (float*)d_out;

    dim3 grid(128, 256);   // (h, b)
    qconv_wmma_kernel<<<grid, 256, 0, stream>>>(x, Ure, Uim, out);
}